// S6_2D_22273700397476
// MI455X (gfx1250) — compile-verified
//
#include <hip/hip_runtime.h>
#include <hip/hip_bf16.h>
#include <math.h>

#if defined(__HIP_DEVICE_COMPILE__) && !__has_builtin(__builtin_amdgcn_wmma_f32_16x16x4_f32)
#error "expected gfx1250 f32 WMMA builtin (device pass)"
#endif

typedef float v2f __attribute__((ext_vector_type(2)));
typedef float v8f __attribute__((ext_vector_type(8)));

#define BB 4
#define HH 48
#define WW 48
#define LL (HH * WW)          // 2304
#define DMODEL 192
#define DSTATE 16
#define EE 384                // EXPAND * DMODEL
#define DTRANK 12
#define MIDC 24               // EE / 16
#define ROWS (BB * LL)        // 9216
#define XDBL_C (DTRANK + 2 * DSTATE)  // 44

__device__ __forceinline__ float silu_f(float x) {
    return x / (1.0f + __expf(-x));
}
__device__ __forceinline__ float softplus_f(float x) {
    return (x > 20.0f) ? x : log1pf(__expf(x));
}

// ---------------------------------------------------------------------------
// C[M,N] = A[M,K] @ W[N,K]^T  via V_WMMA_F32_16X16X4_F32 (fp32 matrix path).
// One wave per 16x16 tile; 8 waves (256 threads) per block.
// M,N multiples of 16; K multiple of 4.
// ---------------------------------------------------------------------------
__global__ __launch_bounds__(256) void gemm_rt_wmma_f32(
    const float* __restrict__ A, const float* __restrict__ W,
    float* __restrict__ C, int M, int N, int K) {
    const int wave = threadIdx.x >> 5;
    const int lane = threadIdx.x & 31;
    const int tilesN = N >> 4;
    const int tile = blockIdx.x * 8 + wave;
    if (tile >= (M >> 4) * tilesN) return;   // wave-uniform: EXEC stays all-1s
    const int tm = (tile / tilesN) << 4;
    const int tn = (tile % tilesN) << 4;
    const int half = lane >> 4;              // K sub-phase: 0 -> K=0,1 ; 1 -> K=2,3
    const int l15 = lane & 15;

    const float* arow = A + (size_t)(tm + l15) * K + 2 * half;  // this lane's A row
    const float* wrow = W + (size_t)(tn + l15) * K + 2 * half;  // this lane's B column

    v8f acc = {};
    for (int k0 = 0; k0 < K; k0 += 4) {
        const float2 af = *(const float2*)(arow + k0);
        const float2 bf = *(const float2*)(wrow + k0);
        v2f a; a.x = af.x; a.y = af.y;
        v2f b; b.x = bf.x; b.y = bf.y;
        acc = __builtin_amdgcn_wmma_f32_16x16x4_f32(
            /*neg_a=*/false, a, /*neg_b=*/false, b,
            /*c_mod=*/(short)0, acc, /*reuse_a=*/false, /*reuse_b=*/false);
    }
    // C/D layout: VGPR r -> M = tm + r + 8*half ; N = tn + (lane&15)
    const int col = tn + l15;
    const int rbase = tm + half * 8;
    #pragma unroll
    for (int r = 0; r < 8; ++r)
        C[(size_t)(rbase + r) * N + col] = acc[r];
}

// ---------------------------------------------------------------------------
// pw1 1x1 conv: hc1[row,m] = pw1_b[m] + sum_e xz[row,e] * pw1_w[m,e]
// (xp = first EE columns of the 2*EE-wide xz buffer)
// ---------------------------------------------------------------------------
__global__ void pw1_kernel(const float* __restrict__ xz,
                           const float* __restrict__ w,
                           const float* __restrict__ bias,
                           float* __restrict__ hc1) {
    int idx = blockIdx.x * blockDim.x + threadIdx.x;
    if (idx >= ROWS * MIDC) return;
    int row = idx / MIDC, m = idx % MIDC;
    const float* xr = xz + (size_t)row * (2 * EE);
    const float* wr = w + (size_t)m * EE;
    float acc = bias[m];
    #pragma unroll 4
    for (int e = 0; e < EE; ++e) acc = fmaf(xr[e], wr[e], acc);
    hc1[idx] = acc;
}

// ---------------------------------------------------------------------------
// depthwise 3x3, zero padding, on (b, h*W+w, m) layout
// ---------------------------------------------------------------------------
__global__ void dwconv_kernel(const float* __restrict__ hc1,
                              const float* __restrict__ dw,
                              float* __restrict__ hc2) {
    int idx = blockIdx.x * blockDim.x + threadIdx.x;
    if (idx >= ROWS * MIDC) return;
    int m = idx % MIDC;
    int row = idx / MIDC;
    int b = row / LL, hw = row % LL;
    int h = hw / WW, w = hw % WW;
    const float* k = dw + m * 9;
    float acc = 0.0f;
    #pragma unroll
    for (int dy = 0; dy < 3; ++dy) {
        int hy = h + dy - 1;
        if (hy < 0 || hy >= HH) continue;
        #pragma unroll
        for (int dx = 0; dx < 3; ++dx) {
            int wx = w + dx - 1;
            if (wx < 0 || wx >= WW) continue;
            acc = fmaf(k[dy * 3 + dx],
                       hc1[((size_t)(b * LL + hy * WW + wx)) * MIDC + m], acc);
        }
    }
    hc2[idx] = acc;
}

// ---------------------------------------------------------------------------
// pw2 1x1 conv + SiLU: x_conv[row,e] = silu( sum_m hc2[row,m] * pw2_w[e,m] )
// ---------------------------------------------------------------------------
__global__ void pw2_silu_kernel(const float* __restrict__ hc2,
                                const float* __restrict__ w,
                                float* __restrict__ x_conv) {
    int idx = blockIdx.x * blockDim.x + threadIdx.x;
    if (idx >= ROWS * EE) return;
    int row = idx / EE, e = idx % EE;
    const float* hr = hc2 + (size_t)row * MIDC;
    const float* wr = w + (size_t)e * MIDC;
    float acc = 0.0f;
    #pragma unroll
    for (int m = 0; m < MIDC; ++m) acc = fmaf(hr[m], wr[m], acc);
    x_conv[idx] = silu_f(acc);
}

// ---------------------------------------------------------------------------
// x_dbl[row,j] = sum_e x_conv[row,e] * x_proj_w[j,e]   (j < 44)
// ---------------------------------------------------------------------------
__global__ void xdbl_kernel(const float* __restrict__ x_conv,
                            const float* __restrict__ w,
                            float* __restrict__ x_dbl) {
    int idx = blockIdx.x * blockDim.x + threadIdx.x;
    if (idx >= ROWS * XDBL_C) return;
    int row = idx / XDBL_C, j = idx % XDBL_C;
    const float* xr = x_conv + (size_t)row * EE;
    const float* wr = w + (size_t)j * EE;
    float acc = 0.0f;
    #pragma unroll 4
    for (int e = 0; e < EE; ++e) acc = fmaf(xr[e], wr[e], acc);
    x_dbl[idx] = acc;
}

// ---------------------------------------------------------------------------
// delta[row,e] = softplus( sum_r x_dbl[row,r]*dt_proj_w[e,r] + 2*dt_proj_b[e] )
// (reference adds the bias once in the matmul, then again before softplus)
// ---------------------------------------------------------------------------
__global__ void delta_kernel(const float* __restrict__ x_dbl,
                             const float* __restrict__ w,
                             const float* __restrict__ bias,
                             float* __restrict__ delta) {
    int idx = blockIdx.x * blockDim.x + threadIdx.x;
    if (idx >= ROWS * EE) return;
    int row = idx / EE, e = idx % EE;
    const float* xr = x_dbl + (size_t)row * XDBL_C;
    const float* wr = w + (size_t)e * DTRANK;
    float acc = 2.0f * bias[e];
    #pragma unroll
    for (int r = 0; r < DTRANK; ++r) acc = fmaf(xr[r], wr[r], acc);
    delta[idx] = softplus_f(acc);
}

// ---------------------------------------------------------------------------
// Selective scan. One block per batch, 384 threads (one per channel e).
// B/C vectors staged through LDS each step. Output written directly in
// final (inverse-permuted) position: at step t, write index order[t].
// ---------------------------------------------------------------------------
__global__ __launch_bounds__(EE) void scan_kernel(
    const float* __restrict__ delta, const float* __restrict__ x_dbl,
    const float* __restrict__ x_conv, const float* __restrict__ A_log,
    const float* __restrict__ D_param, const int* __restrict__ order,
    float* __restrict__ yout) {
    const int b = blockIdx.x;
    const int e = threadIdx.x;
    __shared__ float sB[DSTATE];
    __shared__ float sC[DSTATE];

    float Ae[DSTATE], h[DSTATE];
    #pragma unroll
    for (int n = 0; n < DSTATE; ++n) {
        Ae[n] = -__expf(A_log[e * DSTATE + n]);
        h[n] = 0.0f;
    }
    const float De = D_param[e];

    for (int t = 0; t < LL; ++t) {
        const int p = order[t];
        const size_t rt = (size_t)(b * LL + t);
        const size_t rp = (size_t)(b * LL + p);
        const float d = delta[rt * EE + e];
        const float u = x_conv[rp * EE + e];
        if (e < 2 * DSTATE) {
            float v = x_dbl[rt * XDBL_C + DTRANK + e];
            if (e < DSTATE) sB[e] = v; else sC[e - DSTATE] = v;
        }
        __syncthreads();
        float y = 0.0f;
        #pragma unroll
        for (int n = 0; n < DSTATE; ++n) {
            const float dA = __expf(d * Ae[n]);
            h[n] = fmaf(dA, h[n], d * sB[n] * u);
            y = fmaf(h[n], sC[n], y);
        }
        yout[rp * EE + e] = fmaf(u, De, y);
        __syncthreads();
    }
}

// ---------------------------------------------------------------------------
// gate: g[row,e] = y[row,e] * silu(z[row,e]),  z = columns EE..2EE of xz
// ---------------------------------------------------------------------------
__global__ void gate_kernel(const float* __restrict__ y,
                            const float* __restrict__ xz,
                            float* __restrict__ g) {
    int idx = blockIdx.x * blockDim.x + threadIdx.x;
    if (idx >= ROWS * EE) return;
    int row = idx / EE, e = idx % EE;
    g[idx] = y[idx] * silu_f(xz[(size_t)row * (2 * EE) + EE + e]);
}

// ---------------------------------------------------------------------------
extern "C" void kernel_launch(void* const* d_in, const int* in_sizes, int n_in,
                              void* d_out, int out_size, void* d_ws, size_t ws_size,
                              hipStream_t stream) {
    const float* x         = (const float*)d_in[0];
    const float* in_proj_w = (const float*)d_in[1];
    const float* pw1_w     = (const float*)d_in[2];
    const float* pw1_b     = (const float*)d_in[3];
    const float* dw_w      = (const float*)d_in[4];
    const float* pw2_w     = (const float*)d_in[5];
    const float* x_proj_w  = (const float*)d_in[6];
    const float* dt_proj_w = (const float*)d_in[7];
    const float* dt_proj_b = (const float*)d_in[8];
    const float* A_log     = (const float*)d_in[9];
    const float* D_param   = (const float*)d_in[10];
    const float* out_proj_w= (const float*)d_in[11];
    const int*   order     = (const int*)d_in[12];
    float* out = (float*)d_out;

    // workspace layout (floats)
    float* ws = (float*)d_ws;
    float* xz     = ws;                       // ROWS * 768
    float* x_conv = xz     + (size_t)ROWS * 2 * EE;
    float* hc1    = x_conv + (size_t)ROWS * EE;
    float* hc2    = hc1    + (size_t)ROWS * MIDC;
    float* x_dbl  = hc2    + (size_t)ROWS * MIDC;
    float* delta  = x_dbl  + (size_t)ROWS * XDBL_C;
    float* yscan  = delta  + (size_t)ROWS * EE;
    float* g      = yscan  + (size_t)ROWS * EE;

    const int T1 = ROWS * MIDC;    // 221,184
    const int T2 = ROWS * EE;      // 3,538,944
    const int T3 = ROWS * XDBL_C;  // 405,504

    // 1) xz = x @ in_proj_w^T        (WMMA fp32, M=9216 N=768 K=192)
    {
        int tiles = (ROWS / 16) * (2 * EE / 16);
        gemm_rt_wmma_f32<<<(tiles + 7) / 8, 256, 0, stream>>>(
            x, in_proj_w, xz, ROWS, 2 * EE, DMODEL);
    }
    // 2) pw1 + bias
    pw1_kernel<<<(T1 + 255) / 256, 256, 0, stream>>>(xz, pw1_w, pw1_b, hc1);
    // 3) depthwise 3x3
    dwconv_kernel<<<(T1 + 255) / 256, 256, 0, stream>>>(hc1, dw_w, hc2);
    // 4) pw2 + SiLU
    pw2_silu_kernel<<<(T2 + 255) / 256, 256, 0, stream>>>(hc2, pw2_w, x_conv);
    // 5) x_dbl projection
    xdbl_kernel<<<(T3 + 255) / 256, 256, 0, stream>>>(x_conv, x_proj_w, x_dbl);
    // 6) delta = softplus(dt_proj)
    delta_kernel<<<(T2 + 255) / 256, 256, 0, stream>>>(x_dbl, dt_proj_w, dt_proj_b, delta);
    // 7) selective scan (writes into final positions via order[t])
    scan_kernel<<<BB, EE, 0, stream>>>(delta, x_dbl, x_conv, A_log, D_param, order, yscan);
    // 8) gate with silu(z)
    gate_kernel<<<(T2 + 255) / 256, 256, 0, stream>>>(yscan, xz, g);
    // 9) out = g @ out_proj_w^T      (WMMA fp32, M=9216 N=192 K=384)
    {
        int tiles = (ROWS / 16) * (DMODEL / 16);
        gemm_rt_wmma_f32<<<(tiles + 7) / 8, 256, 0, stream>>>(
            g, out_proj_w, out, ROWS, DMODEL, EE);
    }
}